// GSL_18734647345754
// MI455X (gfx1250) — compile-verified
//
#include <hip/hip_runtime.h>
#include <stdint.h>

// Problem constants (from reference): A is 8192x8192 fp32, K = 64, idx unused.
#define N_COL    8192
#define TOPK     64
#define THREADS  256
#define WAVES    (THREADS / 32)          // 8 wave32s per block
#define V4_PER_T (N_COL / 4 / THREADS)   // 8 float4 per thread

typedef uint32_t u32x4 __attribute__((ext_vector_type(4)));
typedef float    f32x4 __attribute__((ext_vector_type(4)));

// The async-to-LDS b128 builtin wants GCC-vector int4 pointers in explicit
// address spaces (probe: param prints as "int __vector(4) __device__ *").
typedef int gccv4i __attribute__((__vector_size__(16)));
typedef __attribute__((address_space(1))) gccv4i glob_v4i;   // global (64-bit ptr)
typedef __attribute__((address_space(3))) gccv4i lds_v4i;    // LDS (32-bit ptr)

// ---- CDNA5 async global->LDS copy (ASYNCcnt path), with safe fallback ------
__device__ __forceinline__ void async_copy_b128_to_lds(const void* gptr, void* lptr) {
#if defined(__has_builtin)
#if __has_builtin(__builtin_amdgcn_global_load_async_to_lds_b128)
    __builtin_amdgcn_global_load_async_to_lds_b128(
        (glob_v4i*)(uintptr_t)gptr,
        (lds_v4i*)(uint32_t)(uintptr_t)lptr,  // low 32 bits of generic LDS addr == LDS offset
        0 /*imm offset*/, 0 /*cpol*/);
    return;
#endif
#endif
    // Fallback: global_load_b128 + ds_store_b128
    *(u32x4*)lptr = *(const u32x4*)gptr;
}

__device__ __forceinline__ void wait_asynccnt0() {
#if defined(__has_builtin)
#if __has_builtin(__builtin_amdgcn_s_wait_asynccnt)
    __builtin_amdgcn_s_wait_asynccnt(0);
    return;
#endif
#endif
    asm volatile("s_wait_asynccnt 0" ::: "memory");
}

// ---------------------------------------------------------------------------
// One block per row. Stage row (32KB) into LDS once via async global->LDS DMA,
// 4x8-bit radix-select the bit pattern of the 64th-largest relu'd value
// (non-negative IEEE floats order identically as u32), then stream the
// thresholded row out with nontemporal (TH=NT) stores.
//
// Roofline: 512 MB total traffic @ 23.3 TB/s => ~22 us floor. HBM is touched
// exactly once per element in each direction; all selection math is in LDS.
// Zeros (~half the row after relu) are excluded from histogram atomics: they
// can only matter when the threshold is 0, and then the output is full relu
// regardless — the fall-through scan handles that case exactly.
// ---------------------------------------------------------------------------
__global__ __launch_bounds__(THREADS) void GSL_topk_mask_kernel(
    const float* __restrict__ A, float* __restrict__ out, int n_rows) {

    __shared__ __align__(16) uint32_t srow[N_COL];   // raw row bits (32 KB)
    __shared__ uint32_t hist[WAVES][256];            // per-wave32 histograms (8 KB)
    __shared__ uint32_t sel_bin;
    __shared__ uint32_t sel_rem;

    const int    tid  = threadIdx.x;
    const int    wave = tid >> 5;                    // wave32 id
    const size_t row  = blockIdx.x;
    if ((int)row >= n_rows) return;

    const uint32_t* __restrict__ grow = (const uint32_t*)(A + row * (size_t)N_COL);

    // ---- Phase 1: async-stage the row into LDS (HBM read happens exactly once)
    for (int i = 0; i < V4_PER_T; ++i) {
        const int v4 = tid + i * THREADS;
        async_copy_b128_to_lds(grow + (size_t)v4 * 4, &srow[v4 * 4]);
    }
    wait_asynccnt0();
    __syncthreads();

    if (tid == 0) sel_rem = TOPK;
    __syncthreads();

    // ---- Phase 2: 4-pass radix select (MSB->LSB) of the K-th largest key.
    uint32_t prefix = 0;
    for (int shift = 24; shift >= 0; shift -= 8) {
        // zero per-wave histograms (each thread owns bin `tid` of every wave)
        #pragma unroll
        for (int w = 0; w < WAVES; ++w) hist[w][tid] = 0;
        __syncthreads();

        const uint32_t pmask = (shift == 24) ? 0u : (0xFFFFFFFFu << (shift + 8));
        for (int i = 0; i < V4_PER_T; ++i) {
            const int v4 = tid + i * THREADS;
            u32x4 r = *(const u32x4*)&srow[v4 * 4];   // ds_load_b128, conflict-free
            #pragma unroll
            for (int c = 0; c < 4; ++c) {
                float f = __uint_as_float(r[c]);
                f = fmaxf(f, 0.0f);                    // relu
                const uint32_t key = __float_as_uint(f);
                // Skip exact zeros: avoids the bin-0 same-address DS-atomic
                // serialization (~half the row). Correct via fall-through scan.
                if (key != 0u && (key & pmask) == prefix)
                    atomicAdd(&hist[wave][(key >> shift) & 0xFF], 1u);  // ds_add_u32
            }
        }
        __syncthreads();

        // reduce wave-private histograms: thread `tid` owns bin `tid`
        uint32_t s = 0;
        #pragma unroll
        for (int w = 0; w < WAVES; ++w) s += hist[w][tid];
        hist[0][tid] = s;
        __syncthreads();

        // scan bins high->low; bin holding the rem-th largest. Fall-through
        // (cum never reaches rem, e.g. zeros were skipped) lands in bin 0 with
        // the correct residual remaining.
        if (tid == 0) {
            const uint32_t rem = sel_rem;
            uint32_t cum = 0;
            int b = 255;
            for (; b >= 1; --b) {
                const uint32_t c = hist[0][b];
                if (cum + c >= rem) break;
                cum += c;
            }
            sel_bin = (uint32_t)b;          // b==0 on fall-through
            sel_rem = rem - cum;            // remaining rank within chosen bin
        }
        __syncthreads();
        prefix |= sel_bin << shift;
    }

    const uint32_t thr = prefix;  // bit pattern of the K-th largest relu'd value

    // ---- Phase 3: threshold + stream out (write-once -> nontemporal stores)
    float* __restrict__ orow = out + row * (size_t)N_COL;
    for (int i = 0; i < V4_PER_T; ++i) {
        const int v4 = tid + i * THREADS;
        u32x4 r = *(const u32x4*)&srow[v4 * 4];
        f32x4 o;
        #pragma unroll
        for (int c = 0; c < 4; ++c) {
            float f = __uint_as_float(r[c]);
            f = fmaxf(f, 0.0f);
            const uint32_t key = __float_as_uint(f);
            o[c] = (key >= thr) ? f : 0.0f;
        }
        __builtin_nontemporal_store(o, (f32x4*)(orow + (size_t)v4 * 4));
    }
}

extern "C" void kernel_launch(void* const* d_in, const int* in_sizes, int n_in,
                              void* d_out, int out_size, void* d_ws, size_t ws_size,
                              hipStream_t stream) {
    (void)d_ws; (void)ws_size; (void)n_in; (void)in_sizes;
    const float* A = (const float*)d_in[0];
    // d_in[1] (idx, int64) is unused by the reference computation.
    float* out = (float*)d_out;

    const int n_rows = out_size / N_COL;   // 8192
    GSL_topk_mask_kernel<<<dim3(n_rows), dim3(THREADS), 0, stream>>>(A, out, n_rows);
}